// RouteNetOneToOneOutput_25331717112313
// MI455X (gfx1250) — compile-verified
//
#include <hip/hip_runtime.h>

// CDNA5 / gfx1250: wave32, exact-fp32 WMMA (v_wmma_f32_16x16x4_f32) everywhere.
typedef float v2f __attribute__((ext_vector_type(2)));
typedef float v4f __attribute__((ext_vector_type(4)));
typedef float v8f __attribute__((ext_vector_type(8)));

namespace {
constexpr int kNIn  = 784;
constexpr int kNB   = 10;
constexpr int kD    = 10;
constexpr int kC    = kNB * kD;        // 100
constexpr int kNT   = 7;               // N-tiles of 16 for the input GEMM
constexpr int kKQ   = kNIn / 8;        // 98 double-K-steps
constexpr int kWaves = 4;
constexpr int kRowsPerWave  = 16;
constexpr int kRowsPerBlock = kWaves * kRowsPerWave;   // 64
constexpr int kAst = 106;              // LDS acts row stride (even -> b64 aligned)
constexpr int kGst = 106;
constexpr int kActsBufFloats = kWaves * kRowsPerWave * kAst;
constexpr int kSmemFloats = 2 * kActsBufFloats + kWaves * kRowsPerWave * kGst;

// Gate GEMM block-diagonal K-ranges per 16-wide column tile j (rows all <= 99).
constexpr int kJRow0[kNT] = {0, 8, 28, 40, 60, 80, 88};
constexpr int kJCnt [kNT] = {5, 8, 6,  8,  5,  5,  3};

// Packed-fragment workspace layout (floats)
constexpr int kPackWiIds = kNT * kKQ * 32;          // 21952 ids * 4 floats
constexpr int kPackWdIds = 2 * kNB * kNB * 32;      // 6400 ids * 8 floats
constexpr int kPackGIds  = 2 * kNT * 8 * 32;        // 3584 ids * 2 floats
constexpr size_t kWiOff = 0;
constexpr size_t kWdOff = (size_t)kPackWiIds * 4;                 // 87808
constexpr size_t kGOff  = kWdOff + (size_t)kPackWdIds * 8;        // 139008
constexpr int kPackThreads = kPackWiIds + kPackWdIds + kPackGIds; // 31936
} // namespace

__device__ __forceinline__ int imind(int a, int b) { return a < b ? a : b; }

// ---------------------------------------------------------------------------
// One-shot weight pre-swizzle into WMMA fragment order (runs every launch;
// deterministic, ~0.6 MB of d_ws, negligible time).
// ---------------------------------------------------------------------------
__global__ __launch_bounds__(256)
void pack_weights(const float* __restrict__ Wi,   // [100,784]
                  const float* __restrict__ Wg,   // [2,10,10,10]
                  const float* __restrict__ Wd,   // [2,10,10,10,10]
                  float* __restrict__ ws)
{
    const int id = blockIdx.x * 256 + threadIdx.x;
    if (id < kPackWiIds) {
        // B-frag stream for input GEMM: ((nt*98 + kq)*32 + lane)*4
        const int lane = id & 31;
        const int kq   = (id >> 5) % kKQ;
        const int lm = lane & 15, koff = (lane >> 4) * 2;
        const int nt = (id >> 5) / kKQ;
        const int n  = nt * 16 + lm;
        const bool ok = (n < kC);
        const float* src = Wi + (size_t)imind(n, kC - 1) * kNIn + kq * 8 + koff;
        float* dst = ws + kWiOff + (size_t)id * 4;
        dst[0] = ok ? src[0] : 0.f;
        dst[1] = ok ? src[1] : 0.f;
        dst[2] = ok ? src[4] : 0.f;
        dst[3] = ok ? src[5] : 0.f;
    } else if (id < kPackWiIds + kPackWdIds) {
        // B-frags for the data pass: per edge (l,s,t), lane stride 8 floats
        const int id2  = id - kPackWiIds;
        const int lane = id2 & 31;
        const int edge = id2 >> 5;                  // ((l*10+s)*10+t)
        const int lm = lane & 15, koff = (lane >> 4) * 2;
        const float* src = Wd + (size_t)edge * (kD * kD) + imind(lm, kD - 1) * kD;
        float* dst = ws + kWdOff + (size_t)id2 * 8;
        #pragma unroll
        for (int ks = 0; ks < 3; ++ks) {
            const int d0 = ks * 4 + koff;
            const bool ok = (lm < kD) && (d0 < kD); // pairs never straddle the edge
            dst[ks * 2 + 0] = ok ? src[d0]     : 0.f;
            dst[ks * 2 + 1] = ok ? src[d0 + 1] : 0.f;
        }
        dst[6] = 0.f; dst[7] = 0.f;
    } else if (id < kPackThreads) {
        // Block-diagonal gate matrix M1[r=s*10+d, c=s*10+t] = Wg[l,s,t,d]
        const int id3  = id - kPackWiIds - kPackWdIds;
        const int lane = id3 & 31;
        int rest = id3 >> 5;
        const int ks = rest & 7;  rest >>= 3;
        const int j  = rest % kNT;
        const int l  = rest / kNT;
        const int lm = lane & 15, koff = (lane >> 4) * 2;
        const int c = j * 16 + lm;
        const int r = kJRow0[j] + ks * 4 + koff;
        float v0 = 0.f, v1 = 0.f;
        if (c < kC && ks < kJCnt[j]) {
            const int s = c / 10, t = c - s * 10;
            const int d0 = r - s * 10;
            const float* wg = Wg + ((size_t)(l * kNB + s) * kNB + t) * kD;
            if (d0 >= 0 && d0 < kD)     v0 = wg[d0];
            if (d0 + 1 >= 0 && d0 + 1 < kD) v1 = wg[d0 + 1];
        }
        float* dst = ws + kGOff + (size_t)id3 * 2;
        dst[0] = v0; dst[1] = v1;
    }
}

// ---------------------------------------------------------------------------
// Fused RouteNet forward
// ---------------------------------------------------------------------------
__global__ __launch_bounds__(kWaves * 32)
void routenet_fused(const float* __restrict__ x,     // [B, 784]
                    const float* __restrict__ bi,    // [100]
                    const float* __restrict__ bg,    // [2,100]
                    const float* __restrict__ Wout,  // [10,10]
                    const float* __restrict__ pWi,   // packed Wi frags
                    const float* __restrict__ pWd,   // packed Wd frags
                    const float* __restrict__ pG,    // packed gate-matrix frags
                    float* __restrict__ out,         // [B, 10]
                    float* __restrict__ gout)        // [B]
{
    extern __shared__ float smem[];

    const int tid  = threadIdx.x;
    const int wave = tid >> 5;
    const int lane = tid & 31;
    const int lm   = lane & 15;
    const int hi   = lane >> 4;
    const int koff = hi << 1;

    float* actsBuf[2];
    actsBuf[0] = smem;
    actsBuf[1] = smem + kActsBufFloats;
    float* gateBuf = smem + 2 * kActsBufFloats;

    const int row0 = blockIdx.x * kRowsPerBlock + wave * kRowsPerWave;

    // ---------------- Phase A: acts = relu(x @ Wi^T + bi) ----------------
    v8f acc[kNT];
    #pragma unroll
    for (int t = 0; t < kNT; ++t) {
        #pragma unroll
        for (int r = 0; r < 8; ++r) acc[t][r] = 0.f;
    }

    const float* xrow = x + (size_t)(row0 + lm) * kNIn + koff;
    for (int kq = 0; kq < kKQ; ++kq) {             // 98 double-K-steps
        const int kb = kq * 8;
        const v2f a0 = *(const v2f*)(xrow + kb);
        const v2f a1 = *(const v2f*)(xrow + kb + 4);
        #pragma unroll
        for (int nt = 0; nt < kNT; ++nt) {
            const v4f w = *(const v4f*)(pWi + (((size_t)nt * kKQ + kq) * 32 + lane) * 4);
            v2f b0; b0.x = w.x; b0.y = w.y;
            v2f b1; b1.x = w.z; b1.y = w.w;
            acc[nt] = __builtin_amdgcn_wmma_f32_16x16x4_f32(
                false, a0, false, b0, (short)0, acc[nt], false, false);
            acc[nt] = __builtin_amdgcn_wmma_f32_16x16x4_f32(
                false, a1, false, b1, (short)0, acc[nt], false, false);
        }
    }

    {   // bias + relu -> LDS
        float* sAct = actsBuf[0] + (wave * kRowsPerWave) * kAst;
        #pragma unroll
        for (int t = 0; t < kNT; ++t) {
            const int n = t * 16 + lm;
            const bool ok = (n < kC);
            const float bias = ok ? bi[ok ? n : 0] : 0.f;
            #pragma unroll
            for (int r = 0; r < 8; ++r) {
                const int m = r + (hi << 3);
                const float v = fmaxf(acc[t][r] + bias, 0.f);
                if (ok) sAct[m * kAst + n] = v;
            }
        }
    }
    __syncthreads();

    // ---------------- Phase B: two gated layer transitions ----------------
    float gpart = 0.f;
    int cur = 0;
    for (int l = 0; l < 2; ++l) {
        const float* aCur = actsBuf[cur]     + (wave * kRowsPerWave) * kAst;
        float*       aNxt = actsBuf[cur ^ 1] + (wave * kRowsPerWave) * kAst;
        float*       gRow = gateBuf          + (wave * kRowsPerWave) * kGst;
        const float* bgL  = bg + l * kC;

        // ---- gates as a block-diagonal WMMA GEMM: [16x100] @ M1[100x100] ----
        #pragma unroll
        for (int j = 0; j < kNT; ++j) {
            v8f c;
            #pragma unroll
            for (int r = 0; r < 8; ++r) c[r] = 0.f;
            #pragma unroll
            for (int ks = 0; ks < kJCnt[j]; ++ks) {      // rows all <= 99: valid acts
                const int r = kJRow0[j] + ks * 4 + koff;
                const v2f a = *(const v2f*)(aCur + lm * kAst + r);
                const v2f b = *(const v2f*)(pG + (((size_t)(l * kNT + j) * 8 + ks) * 32 + lane) * 2);
                c = __builtin_amdgcn_wmma_f32_16x16x4_f32(
                    false, a, false, b, (short)0, c, false, false);
            }
            const int col = j * 16 + lm;
            const bool okc = (col < kC);
            const float bias = okc ? bgL[okc ? col : 0] : 0.f;
            #pragma unroll
            for (int r = 0; r < 8; ++r) {
                const float gv = fminf(fmaxf(c[r] + bias, 0.f), 1.f);  // hardtanh(.,0,1)
                if (okc) gRow[(r + (hi << 3)) * kGst + col] = gv;
            }
        }
        __syncthreads();

        // gate-sum partial for this row (lane half hi takes st%2==hi edges)
        for (int j = 0; j < 50; ++j)
            gpart += gRow[lm * kGst + ((j << 1) | hi)];

        // ---- data pass: y[:,t,:] = relu( sum_s (g[:,s,t]*a[:,s,:]) @ Wd[l,s,t]^T )
        v8f dacc[kNB];
        #pragma unroll
        for (int t = 0; t < kNB; ++t) {
            #pragma unroll
            for (int r = 0; r < 8; ++r) dacc[t][r] = 0.f;
        }
        for (int s = 0; s < kNB; ++s) {
            const float* aSrc = aCur + lm * kAst + s * kD;
            const v2f av0 = *(const v2f*)(aSrc + koff);       // K rows {0,1}/{2,3}
            const v2f av1 = *(const v2f*)(aSrc + 4 + koff);   // K rows {4,5}/{6,7}
            const v2f t2  = *(const v2f*)(aSrc + 8);          // K rows {8,9} (hi half: pad)
            v2f av2; av2.x = hi ? 0.f : t2.x; av2.y = hi ? 0.f : t2.y;
            const float* wEdge = pWd + ((size_t)((l * kNB + s) * kNB) * 32) * 8;
            #pragma unroll
            for (int t = 0; t < kNB; ++t) {
                const float g = gRow[lm * kGst + s * kNB + t];
                const float* wb = wEdge + ((size_t)t * 32 + lane) * 8;
                const v4f w01 = *(const v4f*)(wb);      // ks0, ks1 frags
                const v2f w2  = *(const v2f*)(wb + 4);  // ks2 frag
                const v2f a0 = av0 * g;                 // v_pk_mul_f32
                const v2f a1 = av1 * g;
                const v2f a2 = av2 * g;
                v2f b0; b0.x = w01.x; b0.y = w01.y;
                v2f b1; b1.x = w01.z; b1.y = w01.w;
                dacc[t] = __builtin_amdgcn_wmma_f32_16x16x4_f32(
                    false, a0, false, b0, (short)0, dacc[t], false, false);
                dacc[t] = __builtin_amdgcn_wmma_f32_16x16x4_f32(
                    false, a1, false, b1, (short)0, dacc[t], false, false);
                dacc[t] = __builtin_amdgcn_wmma_f32_16x16x4_f32(
                    false, a2, false, w2, (short)0, dacc[t], false, false);
            }
        }
        #pragma unroll
        for (int t = 0; t < kNB; ++t) {
            #pragma unroll
            for (int r = 0; r < 8; ++r) {
                const int m = r + (hi << 3);
                if (lm < kD) aNxt[m * kAst + t * kD + lm] = fmaxf(dacc[t][r], 0.f);
            }
        }
        __syncthreads();
        cur ^= 1;
    }

    // ---------------- Phase C: output heads + gate mean ----------------
    const float gtot = gpart + __shfl_xor(gpart, 16, 32);
    if (hi == 0)
        gout[row0 + lm] = gtot * (1.f / 200.f);            // / N_CONN

    const float* aFin = actsBuf[cur] + (wave * kRowsPerWave) * kAst;
    #pragma unroll
    for (int tt = 0; tt < 5; ++tt) {
        const int t = (tt << 1) | hi;
        const float* wo   = Wout + t * kD;
        const float* aRow = aFin + lm * kAst + t * kD;
        float v = 0.f;
        #pragma unroll
        for (int d = 0; d < kD; ++d) v += aRow[d] * wo[d];
        out[(size_t)(row0 + lm) * kNB + t] = fmaxf(v, 0.f);
    }
}

extern "C" void kernel_launch(void* const* d_in, const int* in_sizes, int n_in,
                              void* d_out, int out_size, void* d_ws, size_t ws_size,
                              hipStream_t stream) {
    const float* x    = (const float*)d_in[0];
    const float* Wi   = (const float*)d_in[1];
    const float* bi   = (const float*)d_in[2];
    const float* Wg   = (const float*)d_in[3];
    const float* bg   = (const float*)d_in[4];
    const float* Wd   = (const float*)d_in[5];
    const float* Wout = (const float*)d_in[6];

    float* ws = (float*)d_ws;   // needs ~585 KB of the scratch buffer
    const int B = in_sizes[0] / kNIn;   // 65536
    float* out  = (float*)d_out;
    float* gout = out + (size_t)B * kNB;

    // 1) swizzle weights into WMMA fragment order (deterministic every call)
    pack_weights<<<(kPackThreads + 255) / 256, 256, 0, stream>>>(Wi, Wg, Wd, ws);

    // 2) fused forward pass
    const size_t shmem = (size_t)kSmemFloats * sizeof(float);   // ~79.5 KB
    dim3 grid(B / kRowsPerBlock);
    routenet_fused<<<grid, kWaves * 32, shmem, stream>>>(
        x, bi, bg, Wout, ws + kWiOff, ws + kWdOff, ws + kGOff, out, gout);
}